// SelfAttentionLayer_21148418966055
// MI455X (gfx1250) — compile-verified
//
#include <hip/hip_runtime.h>
#include <hip/hip_bf16.h>

typedef float v2f __attribute__((ext_vector_type(2)));
typedef float v8f __attribute__((ext_vector_type(8)));
typedef unsigned int u32x4 __attribute__((ext_vector_type(4)));
typedef int i32x4 __attribute__((ext_vector_type(4)));
typedef int i32x8 __attribute__((ext_vector_type(8)));

#define DFEAT 2048
#define NKEY  8192
#define DK    64
#define NB    8

// Tensor Data Mover: arity differs between ROCm 7.2 (clang-22, 5 args) and
// therock clang-23 (6 args).
#if defined(__clang_major__) && (__clang_major__ >= 23)
#define TDM_LOAD(g0, g1, z2, z3) \
  __builtin_amdgcn_tensor_load_to_lds((g0), (g1), (z2), (z3), (i32x8){0,0,0,0,0,0,0,0}, 0)
#else
#define TDM_LOAD(g0, g1, z2, z3) \
  __builtin_amdgcn_tensor_load_to_lds((g0), (g1), (z2), (z3), 0)
#endif

// ---------------------------------------------------------------- layernorm
__global__ __launch_bounds__(256) void layernorm_kernel(
    const float* __restrict__ x, const float* __restrict__ g,
    const float* __restrict__ bv, float* __restrict__ y, float eps) {
  __shared__ float red[256];
  const int b = blockIdx.x;
  const int tid = threadIdx.x;
  const float* xr = x + (size_t)b * DFEAT;

  float s = 0.f;
  for (int i = tid; i < DFEAT; i += 256) s += xr[i];
  red[tid] = s; __syncthreads();
  for (int off = 128; off > 0; off >>= 1) {
    if (tid < off) red[tid] += red[tid + off];
    __syncthreads();
  }
  const float mu = red[0] * (1.f / DFEAT);
  __syncthreads();

  float vs = 0.f;
  for (int i = tid; i < DFEAT; i += 256) { float d = xr[i] - mu; vs += d * d; }
  red[tid] = vs; __syncthreads();
  for (int off = 128; off > 0; off >>= 1) {
    if (tid < off) red[tid] += red[tid + off];
    __syncthreads();
  }
  const float var = red[0] * (1.f / DFEAT);
  const float rs = rsqrtf(var + eps);
  for (int i = tid; i < DFEAT; i += 256)
    y[(size_t)b * DFEAT + i] = (xr[i] - mu) * rs * g[i] + bv[i];
}

// ---------------------------------------------------------------- conv for Q
__global__ __launch_bounds__(256) void conv_q_kernel(
    const float* __restrict__ qin, const float* __restrict__ w,
    float* __restrict__ out) {
  __shared__ float ws_[DK * 3];
  if (threadIdx.x < DK * 3) ws_[threadIdx.x] = w[threadIdx.x];
  __syncthreads();
  const int b = blockIdx.y;
  const int x = blockIdx.x * 256 + threadIdx.x;
  const float* xr = qin + (size_t)b * DFEAT;
  const float le = (x > 0) ? xr[x - 1] : 0.f;
  const float mi = xr[x];
  const float ri = (x < DFEAT - 1) ? xr[x + 1] : 0.f;
  float* ob = out + (size_t)b * DK * DFEAT + x;
#pragma unroll
  for (int k = 0; k < DK; ++k)
    ob[(size_t)k * DFEAT] = ws_[3 * k] * le + ws_[3 * k + 1] * mi + ws_[3 * k + 2] * ri;
}

// ------------------------------------------------- fused gather + conv K / V
__global__ __launch_bounds__(256) void conv_kv_kernel(
    const float* __restrict__ feat, const int* __restrict__ idx,
    const float* __restrict__ wk, const float* __restrict__ wv,
    float* __restrict__ keyo, float* __restrict__ vTo) {
  __shared__ float wks[DK * 3], wvs[DK * 3];
  if (threadIdx.x < DK * 3) { wks[threadIdx.x] = wk[threadIdx.x]; wvs[threadIdx.x] = wv[threadIdx.x]; }
  __syncthreads();
  const int b = blockIdx.y;
  const int n = blockIdx.x * 256 + threadIdx.x;
  const float* fr = feat + (size_t)b * DFEAT;
  const float le = (n > 0) ? fr[idx[n - 1]] : 0.f;
  const float mi = fr[idx[n]];
  const float ri = (n < NKEY - 1) ? fr[idx[n + 1]] : 0.f;
  float* kb = keyo + (size_t)b * DK * NKEY + n;
  float* vb = vTo + ((size_t)b * NKEY + n) * DK;
#pragma unroll
  for (int k = 0; k < DK; ++k) {
    kb[(size_t)k * NKEY] = wks[3 * k] * le + wks[3 * k + 1] * mi + wks[3 * k + 2] * ri;
    vb[k]               = wvs[3 * k] * le + wvs[3 * k + 1] * mi + wvs[3 * k + 2] * ri;
  }
}

// ---------------------------------------------------------------- attention
// 1 workgroup = (b, 16 q rows). Keys arrive via TDM double-buffered 64x128
// tiles; 8 waves each take a 16-col chunk of the live window.
__global__ __launch_bounds__(256) void attn_kernel(
    const float* __restrict__ query,   // [8][64][2048]
    const float* __restrict__ key,     // [8][64][8192]
    const float* __restrict__ vT,      // [8][8192][64]
    const float* __restrict__ feat,    // residual [8][2048]
    const float* __restrict__ wconv,   // [64]
    float* __restrict__ attn,          // [8][2048][8192] raw scratch -> final
    float* __restrict__ row_out) {     // [8][2048]
  const int b = blockIdx.y;
  const int qbase = blockIdx.x * 16;
  const int tid = threadIdx.x;
  const int wave = tid >> 5;
  const int lane = tid & 31;
  const int l16 = lane & 15;
  const int half = lane >> 4;   // 0: rows M, K 0/1 ; 1: rows M+8, K 2/3

  __shared__ float ldsK[2][DK][128];   // 64 KB TDM double buffer (key tiles)
  __shared__ float ldsP[8][16][16];    // per-wave P tile re-layout (C -> A)
  __shared__ float ldsO[8][16][64];    // cross-wave O reduction
  __shared__ float ldsM[8][16], ldsL[8][16];
  __shared__ float gM[16], gL[16];

  // A fragments for the whole K=64 (ISA 16x4 f32 A layout)
  v2f afrag[16];
  const float* qb = query + (size_t)b * DK * DFEAT + qbase + l16;
#pragma unroll
  for (int kc = 0; kc < 16; ++kc) {
    const int k0 = 4 * kc + 2 * half;
    afrag[kc].x = qb[(size_t)(k0 + 0) * DFEAT];
    afrag[kc].y = qb[(size_t)(k0 + 1) * DFEAT];
  }

  float m[8], l[8];
#pragma unroll
  for (int i = 0; i < 8; ++i) { m[i] = -3.402823e38f; l[i] = 0.f; }

  const float* kb = key + (size_t)b * DK * NKEY;
  float* attn_b = attn + (size_t)(b * DFEAT + qbase) * NKEY;

  // TDM descriptor: tensor = key[b] (dim0 = 8192 contiguous, dim1 = 64 rows,
  // stride 8192); tile = 128 x 64 fp32 landing row-contiguous in LDS.
  auto tdm_issue = [&](int w, int bufidx) {
    const unsigned long long ga = (unsigned long long)(uintptr_t)(kb + w * 128);
    const unsigned int lds = (unsigned int)(uintptr_t)(&ldsK[bufidx][0][0]);
    u32x4 g0;
    g0[0] = 1u;                                          // count=1, user mode
    g0[1] = lds;                                         // lds_addr [63:32]
    g0[2] = (unsigned int)(ga & 0xffffffffu);            // global_addr lo
    g0[3] = (unsigned int)((ga >> 32) & 0x01ffffffu)     // global_addr hi
          | (2u << 30);                                  // type = 2 (image)
    i32x8 g1;
    g1[0] = 0x00020000;            // wg_mask=0, data_size=2 (4 B)
    g1[1] = (int)(8192u << 16);    // tensor_dim0 lo16 @ [31:16]
    g1[2] = (int)(64u << 16);      // tensor_dim0 hi16=0 | tensor_dim1 lo16
    g1[3] = (int)(128u << 16);     // tensor_dim1 hi16=0 | tile_dim0=128
    g1[4] = 64;                    // tile_dim1=64, tile_dim2=0
    g1[5] = 8192;                  // tensor_dim0_stride lo32
    g1[6] = 0;                     // stride hi16, dim1_stride lo16
    g1[7] = 0;
    const i32x4 z = {0, 0, 0, 0};  // 2D tensor: groups 2/3 unused
    TDM_LOAD(g0, g1, z, z);
  };

  // ---------------- Phase A: scores = Q^T K via TDM-staged key windows
  if (wave == 0) tdm_issue(0, 0);
  for (int w = 0; w < NKEY / 128; ++w) {
    if (wave == 0) __builtin_amdgcn_s_wait_tensorcnt(0);  // window w landed
    __syncthreads();
    if (wave == 0 && w + 1 < NKEY / 128) tdm_issue(w + 1, (w + 1) & 1);
    const int nbase = w * 128 + wave * 16;
    const float (*kt)[128] = ldsK[w & 1];
    // two independent 8-deep WMMA accumulation chains (XDL ILP)
    v8f c0 = {}, c1 = {};
#pragma unroll
    for (int kc = 0; kc < 16; kc += 2) {
      const int ka = 4 * kc + 2 * half;
      const int kb2 = 4 * (kc + 1) + 2 * half;
      v2f bfa, bfb;
      bfa.x = kt[ka + 0][wave * 16 + l16];
      bfa.y = kt[ka + 1][wave * 16 + l16];
      bfb.x = kt[kb2 + 0][wave * 16 + l16];
      bfb.y = kt[kb2 + 1][wave * 16 + l16];
      c0 = __builtin_amdgcn_wmma_f32_16x16x4_f32(false, afrag[kc], false, bfa,
                                                 (short)0, c0, false, false);
      c1 = __builtin_amdgcn_wmma_f32_16x16x4_f32(false, afrag[kc + 1], false, bfb,
                                                 (short)0, c1, false, false);
    }
    const v8f c = c0 + c1;
#pragma unroll
    for (int i = 0; i < 8; ++i) {
      const int row = i + 8 * half;               // C layout: VGPR i -> rows i, i+8
      attn_b[(size_t)row * NKEY + nbase + l16] = c[i];
      float v = c[i];
      v = fmaxf(v, __shfl_xor(v, 1));
      v = fmaxf(v, __shfl_xor(v, 2));
      v = fmaxf(v, __shfl_xor(v, 4));
      v = fmaxf(v, __shfl_xor(v, 8));             // row max within 16-lane half
      const float mn = fmaxf(m[i], v);
      float p = __expf(c[i] - mn);
      p += __shfl_xor(p, 1);
      p += __shfl_xor(p, 2);
      p += __shfl_xor(p, 4);
      p += __shfl_xor(p, 8);                      // row sum within half
      l[i] = l[i] * __expf(m[i] - mn) + p;
      m[i] = mn;
    }
    __syncthreads();                              // release ldsK[w&1] for reuse
  }

  // merge (m, l) across the 8 waves
  if (lane == 0) {
#pragma unroll
    for (int i = 0; i < 8; ++i) { ldsM[wave][i] = m[i]; ldsL[wave][i] = l[i]; }
  } else if (lane == 16) {
#pragma unroll
    for (int i = 0; i < 8; ++i) { ldsM[wave][8 + i] = m[i]; ldsL[wave][8 + i] = l[i]; }
  }
  __syncthreads();
  if (tid < 16) {
    float mm = -3.402823e38f;
    for (int w = 0; w < 8; ++w) mm = fmaxf(mm, ldsM[w][tid]);
    float ll = 0.f;
    for (int w = 0; w < 8; ++w) ll += ldsL[w][tid] * __expf(ldsM[w][tid] - mm);
    gM[tid] = mm; gL[tid] = ll;
  }
  __syncthreads();
  float rm[8], rinv[8];
#pragma unroll
  for (int i = 0; i < 8; ++i) { rm[i] = gM[i + 8 * half]; rinv[i] = 1.f / gL[i + 8 * half]; }

  // ---------------- Phase B: attn = exp(s-m)/l (final write) + O += P V^T
  v8f O[4] = {};
  const float* vb = vT + (size_t)b * NKEY * DK;
  for (int nc = wave; nc < NKEY / 16; nc += 8) {
    const int nbase = nc * 16;
#pragma unroll
    for (int i = 0; i < 8; ++i) {
      const int row = i + 8 * half;
      const size_t ix = (size_t)row * NKEY + nbase + l16;
      const float p = __expf(attn_b[ix] - rm[i]) * rinv[i];
      attn_b[ix] = p;                      // final attention output
      ldsP[wave][row][l16] = p;            // stage for A-layout reload
    }
    asm volatile("s_wait_dscnt 0" ::: "memory");  // wave-local LDS RAW fence
#pragma unroll
    for (int kc = 0; kc < 4; ++kc) {
      v2f pa;                               // A frag: M=l16, K(n)=4kc+2*half(+1)
      pa.x = ldsP[wave][l16][4 * kc + 2 * half];
      pa.y = ldsP[wave][l16][4 * kc + 2 * half + 1];
      const float* vrow = vb + (size_t)(nbase + 4 * kc + 2 * half) * DK;
#pragma unroll
      for (int vt = 0; vt < 4; ++vt) {      // 4 independent accumulator chains
        v2f bv;                             // B frag: rows n, cols v (contiguous)
        bv.x = vrow[vt * 16 + l16];
        bv.y = vrow[DK + vt * 16 + l16];
        O[vt] = __builtin_amdgcn_wmma_f32_16x16x4_f32(false, pa, false, bv,
                                                      (short)0, O[vt], false, false);
      }
    }
  }

  // reduce O across waves, dot with w_conv, add residual
#pragma unroll
  for (int vt = 0; vt < 4; ++vt)
#pragma unroll
    for (int i = 0; i < 8; ++i)
      ldsO[wave][i + 8 * half][vt * 16 + l16] = O[vt][i];
  __syncthreads();
  for (int e = tid; e < 16 * 64; e += 256) {
    const int row = e >> 6, col = e & 63;
    float s = 0.f;
    for (int w = 0; w < 8; ++w) s += ldsO[w][row][col];
    ldsO[0][row][col] = s;
  }
  __syncthreads();
  if (tid < 16) {
    float s = 0.f;
    for (int v = 0; v < DK; ++v) s += ldsO[0][tid][v] * wconv[v];
    const size_t o = (size_t)b * DFEAT + qbase + tid;
    row_out[o] = s + feat[o];
  }
}

// ---------------------------------------------------------------- MLP GEMM
__global__ __launch_bounds__(256) void mlp_kernel(
    const float* __restrict__ x, const float* __restrict__ W,
    const float* __restrict__ bias, const float* __restrict__ res,
    float* __restrict__ out, int relu) {
  __shared__ float xs[NB * DFEAT];  // 64 KB
  for (int i = threadIdx.x; i < NB * DFEAT; i += 256) xs[i] = x[i];
  __syncthreads();
  const int jl = threadIdx.x & 31;
  const int bb = threadIdx.x >> 5;
  const int j = blockIdx.x * 32 + jl;
  const float* wr = W + (size_t)j * DFEAT;
  const float* xr = xs + bb * DFEAT;
  float acc = 0.f;
  for (int i = 0; i < DFEAT; i += 4) {
    acc += wr[i] * xr[i];
    acc += wr[i + 1] * xr[i + 1];
    acc += wr[i + 2] * xr[i + 2];
    acc += wr[i + 3] * xr[i + 3];
  }
  acc += bias[j];
  if (relu) acc = fmaxf(acc, 0.f);
  if (res)  acc += res[(size_t)bb * DFEAT + j];
  out[(size_t)bb * DFEAT + j] = acc;
}

// ---------------------------------------------------------------- launcher
extern "C" void kernel_launch(void* const* d_in, const int* in_sizes, int n_in,
                              void* d_out, int out_size, void* d_ws, size_t ws_size,
                              hipStream_t stream) {
  const float* features = (const float*)d_in[0];
  const int*   sidx     = (const int*)d_in[1];
  const float* w_q      = (const float*)d_in[2];
  const float* w_k      = (const float*)d_in[3];
  const float* w_v      = (const float*)d_in[4];
  const float* w_conv   = (const float*)d_in[5];
  const float* ln1_g    = (const float*)d_in[6];
  const float* ln1_b    = (const float*)d_in[7];
  const float* bn_w1    = (const float*)d_in[8];
  const float* bn_b1    = (const float*)d_in[9];
  const float* bn_w2    = (const float*)d_in[10];
  const float* bn_b2    = (const float*)d_in[11];
  const float* ln2_g    = (const float*)d_in[12];
  const float* ln2_b    = (const float*)d_in[13];

  float* out_vec = (float*)d_out;                       // [8][2048]
  float* attn    = (float*)d_out + NB * DFEAT;          // [8][2048][8192]

  float* ws      = (float*)d_ws;
  float* q_in    = ws;
  float* query   = q_in    + NB * DFEAT;
  float* key     = query   + (size_t)NB * DK * DFEAT;
  float* vT      = key     + (size_t)NB * DK * NKEY;
  float* row_out = vT      + (size_t)NB * NKEY * DK;
  float* x_ln    = row_out + NB * DFEAT;
  float* h1      = x_ln    + NB * DFEAT;

  layernorm_kernel<<<NB, 256, 0, stream>>>(features, ln1_g, ln1_b, q_in, 1e-5f);
  conv_q_kernel <<<dim3(DFEAT / 256, NB), 256, 0, stream>>>(q_in, w_q, query);
  conv_kv_kernel<<<dim3(NKEY / 256, NB), 256, 0, stream>>>(features, sidx, w_k, w_v, key, vT);
  attn_kernel   <<<dim3(DFEAT / 16, NB), 256, 0, stream>>>(query, key, vT, features, w_conv, attn, row_out);
  layernorm_kernel<<<NB, 256, 0, stream>>>(row_out, ln2_g, ln2_b, x_ln, 1e-6f);
  mlp_kernel<<<DFEAT / 32, 256, 0, stream>>>(x_ln, bn_w1, bn_b1, nullptr, h1, 1);
  mlp_kernel<<<DFEAT / 32, 256, 0, stream>>>(h1, bn_w2, bn_b2, row_out, out_vec, 0);
}